// Stage_GNN_learner_74861279969306
// MI455X (gfx1250) — compile-verified
//
#include <hip/hip_runtime.h>

// Forward-value analysis: the pooling stages scale embeddings by
// s + stop_gradient(1-s), which is exactly 1.0f in fp32 RNE, so the output is
//   OUT = Adj @ (relu(Adj @ (X@W1 + b1)) @ W2 + b2)
// Implemented as 4 calls of one fp32 WMMA GEMM kernel (N fixed at 64),
// using V_WMMA_F32_16X16X4_F32 to keep full fp32 precision on the
// uniform-valued adjacency (bf16/f16 WMMA would inject ~0.4% error).

typedef __attribute__((ext_vector_type(2))) float v2f;
typedef __attribute__((ext_vector_type(4))) float v4f;
typedef __attribute__((ext_vector_type(8))) float v8f;
typedef int i32x4 __attribute__((vector_size(16)));

#define NDIM     64
#define MBLK     64
#define KBLK     64
#define LDSTR    68   // padded LDS row stride: 68 mod 64 = 4 -> conflict-free b64 reads
#define NTHREADS 128  // 4 wave32s, one 16-row strip each

#if defined(__gfx1250__) && \
    __has_builtin(__builtin_amdgcn_global_load_async_to_lds_b128) && \
    __has_builtin(__builtin_amdgcn_s_wait_asynccnt)
#define USE_ASYNC 1
#else
#define USE_ASYNC 0
#endif

// exact parameter types per hipcc diagnostic: int4 in AS1 (global) / AS3 (LDS)
typedef __attribute__((address_space(1))) i32x4* gas1p;
typedef __attribute__((address_space(3))) i32x4* gas3p;

__device__ __forceinline__ void copy16_to_lds(const float* g, float* l) {
#if USE_ASYNC
    // flat LDS aperture: addr[31:0] is the LDS byte offset (ISA 10.2)
    __builtin_amdgcn_global_load_async_to_lds_b128(
        (gas1p)(uintptr_t)g, (gas3p)(uintptr_t)l, 0, 0);
#else
    *(v4f*)l = __builtin_nontemporal_load((const v4f*)g);
#endif
}

__device__ __forceinline__ void wait_lds_copies() {
#if USE_ASYNC
    __builtin_amdgcn_s_wait_asynccnt(0);
#endif
}

__global__ __launch_bounds__(NTHREADS) void gemm_n64_wmma_f32(
    const float* __restrict__ A,     // [M, K] row-major
    const float* __restrict__ B,     // [K, 64] row-major
    const float* __restrict__ bias,  // [64] or nullptr
    float* __restrict__ C,           // [M, 64]
    int M, int K, int do_relu)
{
    __shared__ float lA [2][MBLK * LDSTR];  // [m][k], padded
    __shared__ float lBT[2][NDIM * LDSTR];  // [n][k] transposed, padded

    const int tid  = threadIdx.x;
    const int lane = tid & 31;
    const int wave = tid >> 5;          // 0..3
    const int m0   = blockIdx.x * MBLK;

    const int l16 = lane & 15;
    const int hi  = lane >> 4;          // 0 or 1
    const int klo = hi * 2;             // lanes 16..31 carry K+2, K+3

    v8f zero = {};
    v8f acc[4] = {zero, zero, zero, zero};

    const int nT = K / KBLK;

    auto stage = [&](int t, int buf) {
        const int k0 = t * KBLK;
        // A tile: 64x64 f32 = 1024 float4, 8 per thread, coalesced -> LDS (async path)
        #pragma unroll
        for (int i = 0; i < (MBLK * KBLK / 4) / NTHREADS; ++i) {
            int f   = i * NTHREADS + tid;
            int row = f >> 4;                 // 16 float4 per 64-float row
            int c4  = f & 15;
            copy16_to_lds(A + (size_t)(m0 + row) * K + k0 + c4 * 4,
                          &lA[buf][row * LDSTR + c4 * 4]);
        }
        // B tile: load row-major float4, store transposed [n][k] (4 scalar ds stores)
        #pragma unroll
        for (int i = 0; i < (KBLK * NDIM / 4) / NTHREADS; ++i) {
            int f   = i * NTHREADS + tid;
            int kr  = f >> 4;
            int c4  = f & 15;
            v4f v = *(const v4f*)(B + (size_t)(k0 + kr) * NDIM + c4 * 4);
            lBT[buf][(c4 * 4 + 0) * LDSTR + kr] = v.x;
            lBT[buf][(c4 * 4 + 1) * LDSTR + kr] = v.y;
            lBT[buf][(c4 * 4 + 2) * LDSTR + kr] = v.z;
            lBT[buf][(c4 * 4 + 3) * LDSTR + kr] = v.w;
        }
    };

    stage(0, 0);
    wait_lds_copies();
    __syncthreads();

    for (int t = 0; t < nT; ++t) {
        const int buf = t & 1;
        if (t + 1 < nT) stage(t + 1, buf ^ 1);   // overlap next tile with compute

        const float* aRow  = &lA [buf][(wave * 16 + l16) * LDSTR];
        const float* bBase = &lBT[buf][0];
        #pragma unroll
        for (int kk = 0; kk < KBLK; kk += 4) {
            // A fragment 16x4: lanes 0-15 = {K,K+1}, lanes 16-31 = {K+2,K+3}
            v2f av = *(const v2f*)(aRow + kk + klo);
            #pragma unroll
            for (int nt = 0; nt < 4; ++nt) {
                // B fragment 4x16 from transposed tile: contiguous b64 per lane
                v2f bv = *(const v2f*)(bBase + (nt * 16 + l16) * LDSTR + kk + klo);
                acc[nt] = __builtin_amdgcn_wmma_f32_16x16x4_f32(
                    false, av, false, bv, (short)0, acc[nt], false, false);
            }
        }
        if (t + 1 < nT) { wait_lds_copies(); __syncthreads(); }
    }

    // epilogue: 16x16 f32 D layout (lane half -> M+8, VGPR r -> row)
    #pragma unroll
    for (int nt = 0; nt < 4; ++nt) {
        const int col = nt * 16 + l16;
        const float bb = bias ? bias[col] : 0.0f;
        #pragma unroll
        for (int r = 0; r < 8; ++r) {
            int row = wave * 16 + hi * 8 + r;
            float v = acc[nt][r] + bb;
            if (do_relu) v = fmaxf(v, 0.0f);
            C[(size_t)(m0 + row) * NDIM + col] = v;
        }
    }
}

extern "C" void kernel_launch(void* const* d_in, const int* in_sizes, int n_in,
                              void* d_out, int out_size, void* d_ws, size_t ws_size,
                              hipStream_t stream) {
    const float* X   = (const float*)d_in[0];  // features [N, 64]
    const float* Adj = (const float*)d_in[1];  // adj [N, N]
    const float* W1  = (const float*)d_in[2];  // [64, 64]
    const float* b1  = (const float*)d_in[3];  // [64]
    const float* W2  = (const float*)d_in[4];  // [64, 64]
    const float* b2  = (const float*)d_in[5];  // [64]
    // d_in[6] = Ws, d_in[7] = bs: only influence gradients (forward scale == 1.0f)
    float* out = (float*)d_out;

    const int N = in_sizes[0] / NDIM;          // 8192

    float* T1 = (float*)d_ws;                  // [N, 64]
    float* H  = T1 + (size_t)N * NDIM;         // [N, 64]
    float* T2 = H  + (size_t)N * NDIM;         // [N, 64]

    dim3 blk(NTHREADS);
    dim3 grd(N / MBLK);                        // 128 workgroups per GEMM

    // T1 = X @ W1 + b1
    gemm_n64_wmma_f32<<<grd, blk, 0, stream>>>(X,   W1, b1,      T1,  N, NDIM, 0);
    // H = relu(Adj @ T1)
    gemm_n64_wmma_f32<<<grd, blk, 0, stream>>>(Adj, T1, nullptr, H,   N, N,    1);
    // T2 = H @ W2 + b2
    gemm_n64_wmma_f32<<<grd, blk, 0, stream>>>(H,   W2, b2,      T2,  N, NDIM, 0);
    // OUT = Adj @ T2
    gemm_n64_wmma_f32<<<grd, blk, 0, stream>>>(Adj, T2, nullptr, out, N, N,    0);
}